// Mapper_50105088475226
// MI455X (gfx1250) — compile-verified
//
#include <hip/hip_runtime.h>
#include <cstdint>

// FCOS target-assignment mapper for MI455X (gfx1250).
// Bandwidth-bound (~60 MB of stores, negligible input/compute), so the design
// goals are: coalesced channel-planar stores, LDS-resident box table staged via
// the CDNA5 async global->LDS DMA path, one pass per level.

#define NBOX 64
#define NB   8
#define NCLS 81
#define NCH  86   // 4 reg + 1 centerness + 81 cls

#ifndef __has_builtin
#define __has_builtin(x) 0
#endif

#if defined(__HIP_DEVICE_COMPILE__) &&                                   \
    __has_builtin(__builtin_amdgcn_global_load_async_to_lds_b128) &&     \
    __has_builtin(__builtin_amdgcn_global_load_async_to_lds_b32)
#define USE_ASYNC_LDS 1
#else
#define USE_ASYNC_LDS 0
#endif

#if USE_ASYNC_LDS
// Builtin signatures (from clang diagnostics / BuiltinsAMDGPU.def):
//   b128: void(v4i addrspace(1)*, v4i addrspace(3)*, imm int, imm int)
//   b32 : void(int addrspace(1)*, int addrspace(3)*, imm int, imm int)
typedef int v4i __attribute__((__vector_size__(16)));
typedef __attribute__((address_space(1))) v4i* as1_v4i;
typedef __attribute__((address_space(3))) v4i* as3_v4i;
typedef __attribute__((address_space(1))) int* as1_i32;
typedef __attribute__((address_space(3))) int* as3_i32;

__device__ __forceinline__ as1_v4i g_as1_v4i(const void* p) {
  // global generic pointer: numeric value == global address
  return (as1_v4i)(uintptr_t)p;
}
__device__ __forceinline__ as3_v4i l_as3_v4i(const void* p) {
  // generic LDS pointer: low 32 bits are the LDS byte offset (flat aperture)
  return (as3_v4i)(uint32_t)(uintptr_t)p;
}
__device__ __forceinline__ as1_i32 g_as1_i32(const void* p) {
  return (as1_i32)(uintptr_t)p;
}
__device__ __forceinline__ as3_i32 l_as3_i32(const void* p) {
  return (as3_i32)(uint32_t)(uintptr_t)p;
}
#endif

__device__ __forceinline__ void wait_async_zero() {
#if __has_builtin(__builtin_amdgcn_s_wait_asynccnt)
  __builtin_amdgcn_s_wait_asynccnt(0);
#else
  asm volatile("s_wait_asynccnt 0" ::: "memory");
#endif
}

// One block covers a contiguous run of pixels (x fastest) of one batch image at
// one pyramid level. Steps:
//  1) async-DMA the 64 raw boxes (float4) + labels of this batch into LDS
//  2) stable descending-area rank sort inside LDS (64 elements, O(N^2) ranks)
//  3) each thread resolves its pixel's winning box and writes 86 channels,
//     each store coalesced across the wave (consecutive x).
__global__ void fcos_map_level_kernel(const float*  __restrict__ boxes,
                                      const int*    __restrict__ labels,
                                      float*        __restrict__ out,
                                      int H, float ps, float th0, float th1) {
  const int b   = blockIdx.y;
  const int tid = threadIdx.x;

  __shared__ float4 raw[NBOX];     // unsorted boxes (DMA target)
  __shared__ int    rawl[NBOX];    // unsorted labels (DMA target)
  __shared__ float  area[NBOX];
  __shared__ float4 sb[NBOX];      // boxes sorted by descending area (stable)
  __shared__ int    sl[NBOX];      // labels in same order

  // ---- stage raw boxes/labels into LDS ----
#if USE_ASYNC_LDS
  if (tid < NBOX) {
    const float4* gsrc = (const float4*)(boxes) + (size_t)b * NBOX + tid;
    const int*    lsrc = labels + (size_t)b * NBOX + tid;
    __builtin_amdgcn_global_load_async_to_lds_b128(g_as1_v4i(gsrc),
                                                   l_as3_v4i(&raw[tid]), 0, 0);
    __builtin_amdgcn_global_load_async_to_lds_b32 (g_as1_i32(lsrc),
                                                   l_as3_i32(&rawl[tid]), 0, 0);
  }
  wait_async_zero();
#else
  if (tid < NBOX) {
    const float4* gsrc = (const float4*)(boxes) + (size_t)b * NBOX + tid;
    raw[tid]  = *gsrc;
    rawl[tid] = labels[(size_t)b * NBOX + tid];
  }
#endif
  __syncthreads();

  // ---- stable descending-area sort via rank computation ----
  if (tid < NBOX) {
    const float4 bb = raw[tid];
    area[tid] = (bb.z - bb.x) * (bb.w - bb.y);
  }
  __syncthreads();
  if (tid < NBOX) {
    const float a = area[tid];
    int rank = 0;
    #pragma unroll
    for (int j = 0; j < NBOX; ++j) {
      const float aj = area[j];
      // jnp.argsort(-area) is stable: j precedes i iff aj > a, or tie && j < i
      rank += (aj > a || (aj == a && j < tid)) ? 1 : 0;
    }
    sb[rank] = raw[tid];
    sl[rank] = rawl[tid];
  }
  __syncthreads();

  // ---- per-pixel assignment ----
  const int HH  = H * H;
  const int pix = blockIdx.x * blockDim.x + tid;
  if (pix >= HH) return;   // never taken for our block sizes (HH % block == 0)

  const int   y  = pix / H;
  const int   x  = pix - y * H;
  const float mx = ((float)x + 0.5f) * ps;
  const float my = ((float)y + 0.5f) * ps;

  int  win    = -1;
  bool fg_any = false;
  for (int n = 0; n < NBOX; ++n) {
    const float4 bb = sb[n];
    const float l = mx - bb.x;
    const float t = my - bb.y;
    const float r = bb.z - mx;
    const float d = bb.w - my;
    const float mn = fminf(fminf(l, t), fminf(r, d));
    const float mv = fmaxf(fmaxf(l, t), fmaxf(r, d));
    const bool  fg = (mn >= 0.0f);
    fg_any |= fg;
    if (fg && (mv > th0) && (mv <= th1)) win = n;  // max index == smallest area
  }

  float r0 = 0.0f, r1 = 0.0f, r2 = 0.0f, r3 = 0.0f, cent = 0.0f;
  int lab = -1;
  if (win >= 0) {
    const float4 bw = sb[win];
    r0 = mx - bw.x;
    r1 = my - bw.y;
    r2 = bw.z - mx;
    r3 = bw.w - my;
    const float dx = fminf(r0, r2), Dx = fmaxf(r0, r2);
    const float dy = fminf(r1, r3), Dy = fmaxf(r1, r3);
    const float arg = (dx / ((Dx != 0.0f) ? Dx : 1.0f)) *
                      (dy / ((Dy != 0.0f) ? Dy : 1.0f));
    cent = (arg > 0.0f) ? sqrtf(arg) : 0.0f;
    lab  = sl[win];
  }

  // out layout: [b][c][y][x]; lanes of a wave cover consecutive x -> every
  // per-channel store below is a fully coalesced contiguous burst.
  float* o = out + (size_t)b * NCH * (size_t)HH + (size_t)pix;
  o[0]              = r0;
  o[(size_t)HH]     = r1;
  o[2 * (size_t)HH] = r2;
  o[3 * (size_t)HH] = r3;
  o[4 * (size_t)HH] = cent;

  const float bg0 = (win < 0 && !fg_any) ? 1.0f : 0.0f;
  #pragma unroll
  for (int k = 0; k < NCLS; ++k) {
    const float v = (win >= 0) ? ((k == lab) ? 1.0f : 0.0f)
                               : ((k == 0) ? bg0 : 0.0f);
    o[(size_t)(5 + k) * HH] = v;
  }
}

extern "C" void kernel_launch(void* const* d_in, const int* in_sizes, int n_in,
                              void* d_out, int out_size, void* d_ws, size_t ws_size,
                              hipStream_t stream) {
  (void)in_sizes; (void)n_in; (void)out_size; (void)d_ws; (void)ws_size;
  const float* boxes  = (const float*)d_in[0];   // (8, 64, 4) f32
  const int*   labels = (const int*)d_in[1];     // (8, 64)    i32
  float*       out    = (float*)d_out;

  static const int   strides[5] = {8, 16, 32, 64, 128};
  // LEVEL_TH from _calc_level_thresholds(STRIDES, 1024):
  static const float th0s[5] = {0.0078125f, 0.0625f, 0.125f, 0.25f, 0.5f};
  static const float th1s[5] = {0.0625f,    0.125f,  0.25f,  0.5f,  1.0f};

  size_t off = 0;
  for (int lv = 0; lv < 5; ++lv) {
    const int   H  = 1024 / strides[lv];
    const int   HH = H * H;
    const float ps = (float)strides[lv] / 1024.0f;
    const int   block = (HH < 256) ? HH : 256;       // HH % block == 0 always
    dim3 grid((HH + block - 1) / block, NB);
    fcos_map_level_kernel<<<grid, dim3(block), 0, stream>>>(
        boxes, labels, out + off, H, ps, th0s[lv], th1s[lv]);
    off += (size_t)NB * NCH * (size_t)HH;
  }
}